// bitshift_codebook_403726926457
// MI455X (gfx1250) — compile-verified
//
#include <hip/hip_runtime.h>

// out[v, t, n] = tlut[encoded[t, n], v]
//   tlut:    (65536, 2) float32  -> array of float2 (8B {v0,v1} per entry)
//   encoded: (TB*N,) int32 flat  -> streamed once (NT)
//   out:     (2, TB*N) float32   -> two coalesced planes, streamed once (NT)
//
// Pure gather; HBM-bound: 128 MB idx read + 256 MB store ~= 384 MB
// -> ~16.5 us floor @ 23.3 TB/s. The 512 KB LUT stays resident in the
// 192 MB L2; NT hints keep the 384 MB of streaming traffic from evicting it.
//
// 8 elements / thread = two independent b128 index loads per lane, so each
// wave keeps 2 index loads + 8 gathers in flight instead of serializing on a
// single loadcnt==0 wait.

typedef int   i32x4 __attribute__((ext_vector_type(4)));
typedef float f32x4 __attribute__((ext_vector_type(4)));
typedef float f32x2 __attribute__((ext_vector_type(2)));

__global__ __launch_bounds__(256) void bitshift_codebook_dequant_kernel(
    const f32x2* __restrict__ lut,   // 2^16 entries {v0, v1}
    const int*   __restrict__ enc,   // total int32 indices
    float*       __restrict__ out,   // 2 * total floats
    int total)
{
    int i = (blockIdx.x * blockDim.x + threadIdx.x) * 8;
    if (i >= total) return;

    // Warm L2 ahead of the demand stream (global_prefetch_b8, DEV scope).
    __builtin_prefetch(enc + i + 32768, 0, 1);

    // Two independent coalesced b128 NT loads: 8 indices in flight at once.
    i32x4 eA = __builtin_nontemporal_load((const i32x4*)(enc + i));
    i32x4 eB = __builtin_nontemporal_load((const i32x4*)(enc + i + 4));

    // Eight b64 gathers from the L2-resident LUT (default RT hint: keep hot).
    f32x2 a0 = lut[eA.x];
    f32x2 a1 = lut[eA.y];
    f32x2 a2 = lut[eA.z];
    f32x2 a3 = lut[eA.w];
    f32x2 b0 = lut[eB.x];
    f32x2 b1 = lut[eB.y];
    f32x2 b2 = lut[eB.z];
    f32x2 b3 = lut[eB.w];

    f32x4 p0 = { a0.x, a1.x, a2.x, a3.x };   // plane v=0, first 4
    f32x4 p1 = { b0.x, b1.x, b2.x, b3.x };   // plane v=0, next 4
    f32x4 q0 = { a0.y, a1.y, a2.y, a3.y };   // plane v=1, first 4
    f32x4 q1 = { b0.y, b1.y, b2.y, b3.y };   // plane v=1, next 4

    // Four coalesced b128 NT stores (streamed, never re-read).
    __builtin_nontemporal_store(p0, (f32x4*)(out + i));
    __builtin_nontemporal_store(p1, (f32x4*)(out + i + 4));
    __builtin_nontemporal_store(q0, (f32x4*)(out + total + i));
    __builtin_nontemporal_store(q1, (f32x4*)(out + total + i + 4));
}

extern "C" void kernel_launch(void* const* d_in, const int* in_sizes, int n_in,
                              void* d_out, int out_size, void* d_ws, size_t ws_size,
                              hipStream_t stream) {
    const f32x2* lut = (const f32x2*)d_in[0];   // tlut: (65536, 2) f32
    const int*   enc = (const int*)d_in[1];     // encoded: TB*N int32
    float*       out = (float*)d_out;           // (2, TB*N) f32

    int total = in_sizes[1];                    // TB * N = 33,554,432
    const int threads   = 256;
    const int per_block = threads * 8;
    int blocks = (total + per_block - 1) / per_block;

    bitshift_codebook_dequant_kernel<<<blocks, threads, 0, stream>>>(
        lut, enc, out, total);
}